// CapsuleLayer_61899068670106
// MI455X (gfx1250) — compile-verified
//
#include <hip/hip_runtime.h>
#include <math.h>

// ---------------- CDNA5 capsule routing, fused WMMA pipeline ----------------
// u[b,i,j,d] = sum_e W[i,j,d,e] * x[b,i,e]  -> per-i GEMM M=64,N=512,K=16
// u is never materialized: recomputed per routing pass with
// v_wmma_f32_16x16x4_f32. W_i/X_i are staged to LDS with double-buffered
// ASYNC global->LDS copies (ASYNCcnt), overlapping HBM streaming with WMMA.

typedef __attribute__((ext_vector_type(2))) float v2f;
typedef __attribute__((ext_vector_type(8))) float v8f;
typedef __attribute__((ext_vector_type(4))) int   v4i;

#define N_IN   2048
#define JCAP   32
#define DCAP   16
#define BATCH  64
#define WPITCH 20      // padded LDS row pitch in floats (bank-safe, 16B aligned)
#define LPITCH 33      // logit LDS row pitch (odd -> conflict-free softmax)
#define IPB    16      // i's per workgroup
#define EPS_SQ 1e-8f

#define WBUF   (512 * WPITCH)            // one W_i LDS buffer, floats
#define XBUF   (64 * WPITCH)             // one X_i LDS buffer, floats
#define SMEM_FLOATS (2 * WBUF + 2 * XBUF + 64 * LPITCH)
#define SMEM_BYTES  (SMEM_FLOATS * 4)    // 100608 B < 320 KB WGP LDS

#if __has_builtin(__builtin_amdgcn_global_load_async_to_lds_b128)
#define HAVE_ASYNC_LDS_BUILTIN 1
#endif
#if __has_builtin(__builtin_amdgcn_s_wait_asynccnt)
#define HAVE_WAIT_ASYNC_BUILTIN 1
#endif

__device__ __forceinline__ void async_copy_b128(const float* gsrc, float* ldst) {
#ifdef HAVE_ASYNC_LDS_BUILTIN
    __builtin_amdgcn_global_load_async_to_lds_b128(
        (__attribute__((address_space(1))) v4i*)(void*)gsrc,
        (__attribute__((address_space(3))) v4i*)(void*)ldst,
        0 /*offset*/, 0 /*cpol*/);
#else
    unsigned lds32 = (unsigned)(unsigned long long)ldst;   // generic lo32 == LDS addr
    asm volatile("global_load_async_to_lds_b128 %0, %1, off"
                 :: "v"(lds32), "v"((unsigned long long)gsrc) : "memory");
#endif
}

template <int N>
__device__ __forceinline__ void wait_asynccnt() {
#ifdef HAVE_WAIT_ASYNC_BUILTIN
    __builtin_amdgcn_s_wait_asynccnt(N);
#else
    asm volatile("s_wait_asynccnt %0" :: "i"(N) : "memory");
#endif
}

// Issue exactly 5 async-to-LDS b128 instructions per thread (uniform per wave):
// 4 cover W_i (512x16 f32, pitch-padded rows), 1 covers X_i (64x16 f32; lanes
// 256..511 duplicate lanes 0..255 -> identical benign writes, keeps ASYNCcnt
// accounting uniform at 5 per wave per buffer).
__device__ __forceinline__ void stage_async(float* __restrict__ Wbuf,
                                            float* __restrict__ Xbuf,
                                            const float* __restrict__ w,
                                            const float* __restrict__ x,
                                            int i, int tid)
{
    const float* gw = w + (size_t)i * 8192 + (size_t)tid * 16;
    float* lw = Wbuf + tid * WPITCH;
#pragma unroll
    for (int q = 0; q < 4; ++q)
        async_copy_b128(gw + q * 4, lw + q * 4);

    const int chunk = tid & 255;                     // 256 x 16B chunks of X_i
    const float* gx = x + (size_t)(chunk >> 2) * (N_IN * 16)
                        + (size_t)i * 16 + (chunk & 3) * 4;
    float* lx = Xbuf + (chunk >> 2) * WPITCH + (chunk & 3) * 4;
    async_copy_b128(gx, lx);
}

__global__ void caps_zero(float* __restrict__ p, int n) {
    int i = blockIdx.x * blockDim.x + threadIdx.x;
    if (i < n) p[i] = 0.0f;
}

// One workgroup = 512 threads = 16 waves. Wave w: mtile = w%4 (16 batch rows),
// jbase = (w/4)*8 (8 capsules). Each wave holds 8 u-tiles + 8 s-accumulator
// tiles in VGPRs across the IPB-deep i loop; s is flushed once via f32 atomics.
template <bool ROUTE>
__global__ __launch_bounds__(512)
void caps_pass(const float* __restrict__ x,      // [64, 2048, 1, 16]
               const float* __restrict__ w,      // [2048, 32, 16, 16]
               const float* __restrict__ v,      // [64, 32, 16] (ROUTE only)
               float* __restrict__ s_out)        // [64, 32, 16] accumulated
{
    extern __shared__ float smem[];
    float* Wl = smem;                  // [2][512*WPITCH]
    float* Xl = Wl + 2 * WBUF;         // [2][64*WPITCH]
    float* Ll = Xl + 2 * XBUF;         // [64*LPITCH] logits / softmax c

    const int tid   = threadIdx.x;
    const int lane  = tid & 31;
    const int wave  = tid >> 5;          // 0..15
    const int mtile = wave & 3;          // batch tile: b in [mtile*16, +16)
    const int jbase = (wave >> 2) * 8;   // capsule range: j in [jbase, +8)
    const int half  = lane >> 4;         // 0 / 1
    const int dl    = lane & 15;         // d = column within 16x16 tile

    v8f sacc[8];
#pragma unroll
    for (int t = 0; t < 8; ++t) sacc[t] = (v8f)0.0f;

    const int i0 = blockIdx.x * IPB;

    // prologue: async-stage buffer 0
    stage_async(Wl, Xl, w, x, i0, tid);

#pragma unroll 1
    for (int ii = 0; ii < IPB; ++ii) {
        const int cur = ii & 1;
        float* Wc = Wl + cur * WBUF;
        float* Xc = Xl + cur * XBUF;

        // prefetch next buffer, then retire the current one (in-order ASYNCcnt)
        if (ii + 1 < IPB) {
            const int nxt = cur ^ 1;
            stage_async(Wl + nxt * WBUF, Xl + nxt * XBUF, w, x, i0 + ii + 1, tid);
            wait_asynccnt<5>();
        } else {
            wait_asynccnt<0>();
        }
        __syncthreads();

        // ---- A fragments (X tile, 16x4 per k-step): lanes 0-15 k={0,1}, 16-31 k={2,3} ----
        v2f af[4];
#pragma unroll
        for (int s = 0; s < 4; ++s)
            af[s] = *(const v2f*)(Xc + (mtile * 16 + dl) * WPITCH + s * 4 + half * 2);

        // ---- u tiles: D = A x B, B[k][n] = W[n][k] ----
        v8f u[8];
#pragma unroll
        for (int t = 0; t < 8; ++t) {
            v8f acc = (v8f)0.0f;
            const int n = (jbase + t) * 16 + dl;     // row of W_i = (j, d)
#pragma unroll
            for (int s = 0; s < 4; ++s) {
                v2f bf = *(const v2f*)(Wc + n * WPITCH + s * 4 + half * 2);
                acc = __builtin_amdgcn_wmma_f32_16x16x4_f32(
                    false, af[s], false, bf, (short)0, acc, false, false);
            }
            u[t] = acc;
        }

        if (!ROUTE) {
            // first iteration: c = softmax(0) = 1/J uniform; scale applied later
#pragma unroll
            for (int t = 0; t < 8; ++t) sacc[t] += u[t];
            __syncthreads();
        } else {
            // ---- logits[b,i,j] = sum_d u[b,i,j,d] * v[b,j,d] ----
#pragma unroll
            for (int t = 0; t < 8; ++t) {
                const int j = jbase + t;
#pragma unroll
                for (int r = 0; r < 8; ++r) {
                    const int b = mtile * 16 + half * 8 + r;
                    float p = u[t][r] * v[b * 512 + j * 16 + dl];
                    p += __shfl_xor(p, 1);
                    p += __shfl_xor(p, 2);
                    p += __shfl_xor(p, 4);
                    p += __shfl_xor(p, 8);
                    if (dl == 0) Ll[b * LPITCH + j] = p;
                }
            }
            __syncthreads();
            // ---- softmax over j (local per (b,i)); threads 0..63 own one b ----
            if (tid < 64) {
                float m = -1e30f;
#pragma unroll
                for (int jj = 0; jj < JCAP; ++jj) m = fmaxf(m, Ll[tid * LPITCH + jj]);
                float sum = 0.0f;
#pragma unroll
                for (int jj = 0; jj < JCAP; ++jj) {
                    float e = __expf(Ll[tid * LPITCH + jj] - m);
                    Ll[tid * LPITCH + jj] = e;
                    sum += e;
                }
                float inv = 1.0f / sum;
#pragma unroll
                for (int jj = 0; jj < JCAP; ++jj) Ll[tid * LPITCH + jj] *= inv;
            }
            __syncthreads();
            // ---- s += c * u ----
#pragma unroll
            for (int t = 0; t < 8; ++t) {
                const int j = jbase + t;
#pragma unroll
                for (int r = 0; r < 8; ++r) {
                    const int b = mtile * 16 + half * 8 + r;
                    sacc[t][r] += Ll[b * LPITCH + j] * u[t][r];
                }
            }
            __syncthreads();
        }
    }

    // ---- flush partial s via fp32 global atomics (global_atomic_add_f32) ----
#pragma unroll
    for (int t = 0; t < 8; ++t) {
        const int j = jbase + t;
#pragma unroll
        for (int r = 0; r < 8; ++r) {
            const int b = mtile * 16 + half * 8 + r;
            unsafeAtomicAdd(&s_out[b * 512 + j * 16 + dl], sacc[t][r]);
        }
    }
}

// squash over capsule axis J (reference's axis=2 quirk): sq per (b,d) = sum_j s^2
// out = addv + squash(s*scale)  (addv=null -> just squash)
__global__ __launch_bounds__(512)
void caps_squash_j(const float* __restrict__ s, float scale,
                   const float* __restrict__ addv, float* __restrict__ out)
{
    __shared__ float sh[512];
    __shared__ float sq16[16];
    const int b = blockIdx.x, tid = threadIdx.x;
    float val = s[b * 512 + tid] * scale;
    sh[tid] = val * val;
    __syncthreads();
    if (tid < 16) {
        float a = 0.0f;
        for (int j = 0; j < JCAP; ++j) a += sh[j * 16 + tid];
        sq16[tid] = a;
    }
    __syncthreads();
    float sq  = sq16[tid & 15];
    float res = val * (sq / (1.0f + sq)) / sqrtf(sq + EPS_SQ);
    if (addv) res += addv[b * 512 + tid];
    out[b * 512 + tid] = res;
}

// final squash over D (axis=-1): sq per (b,j) = sum_d s^2
__global__ __launch_bounds__(512)
void caps_squash_d(const float* __restrict__ s, float* __restrict__ out)
{
    __shared__ float sh[512];
    __shared__ float sq32[32];
    const int b = blockIdx.x, tid = threadIdx.x;
    float val = s[b * 512 + tid];
    sh[tid] = val * val;
    __syncthreads();
    if (tid < 32) {
        float a = 0.0f;
        for (int d = 0; d < DCAP; ++d) a += sh[tid * 16 + d];
        sq32[tid] = a;
    }
    __syncthreads();
    float sq = sq32[tid >> 4];
    out[b * 512 + tid] = val * (sq / (1.0f + sq)) / sqrtf(sq + EPS_SQ);
}

extern "C" void kernel_launch(void* const* d_in, const int* in_sizes, int n_in,
                              void* d_out, int out_size, void* d_ws, size_t ws_size,
                              hipStream_t stream)
{
    const float* x = (const float*)d_in[0];   // [64,2048,1,16]
    const float* w = (const float*)d_in[1];   // [1,2048,32,16,16]
    float* out = (float*)d_out;               // [64,32,16]

    float* ws   = (float*)d_ws;               // 5 x 32768 floats = 640 KB
    float* s0   = ws;
    float* s1   = ws + 32768;
    float* s2   = ws + 65536;
    float* v1   = ws + 98304;
    float* vsum = ws + 131072;

    const int SN = 64 * JCAP * DCAP;          // 32768

    // zero the three atomic accumulators (ws is poisoned)
    caps_zero<<<(3 * SN + 255) / 256, 256, 0, stream>>>(s0, 3 * SN);

    // iter 1: c uniform -> s0 = sum_i u ; v1 = squash_J(s0 / J)
    caps_pass<false><<<N_IN / IPB, 512, SMEM_BYTES, stream>>>(x, w, nullptr, s0);
    caps_squash_j<<<BATCH, 512, 0, stream>>>(s0, 1.0f / (float)JCAP, nullptr, v1);

    // iter 2: b1 = sum_d u*v1 ; c = softmax_j(b1) ; s1 = sum_i c*u
    caps_pass<true><<<N_IN / IPB, 512, SMEM_BYTES, stream>>>(x, w, v1, s1);
    // vsum = v1 + squash_J(s1)   (b2 = sum_d u*(v1+v2))
    caps_squash_j<<<BATCH, 512, 0, stream>>>(s1, 1.0f, v1, vsum);

    // final: c = softmax_j(sum_d u*(v1+v2)) ; s2 = sum_i c*u ; out = squash_D(s2)
    caps_pass<true><<<N_IN / IPB, 512, SMEM_BYTES, stream>>>(x, w, vsum, s2);
    caps_squash_d<<<BATCH, 512, 0, stream>>>(s2, out);
}